// PINN_41549513621995
// MI455X (gfx1250) — compile-verified
//
#include <hip/hip_runtime.h>

// ---------------------------------------------------------------------------
// CDNA5 (gfx1250) PINN loss. fp32 WMMA (16x16x4) everywhere matmuls occur.
// Inner loops are explicitly software-pipelined: fragment loads for step k+1
// are issued before the WMMAs of step k, so s_wait_dscnt never drains to 0
// inside the loop and the matrix pipe stays busy.
// ---------------------------------------------------------------------------

typedef float v2f __attribute__((ext_vector_type(2)));
typedef float v4f __attribute__((ext_vector_type(4)));
typedef float v8f __attribute__((ext_vector_type(8)));

__device__ __forceinline__ v8f wmma4(v2f a, v2f b, v8f c) {
  // D = A(16x4) * B(4x16) + C(16x16), all fp32, wave32
  return __builtin_amdgcn_wmma_f32_16x16x4_f32(false, a, false, b, (short)0, c,
                                               false, false);
}

__device__ __forceinline__ v8f splat8(float x) {
  v8f v = {x, x, x, x, x, x, x, x};
  return v;
}

__device__ __forceinline__ float tanh_fast(float x) {
  // tanh(x) = 1 - 2/(e^{2x}+1); v_exp_f32 + v_rcp_f32, saturates correctly
  float e = __builtin_amdgcn_exp2f(x * 2.8853900817779268f);
  return 1.0f - 2.0f * __builtin_amdgcn_rcpf(e + 1.0f);
}

// ---- workspace layout (floats) --------------------------------------------
#define WS_ACC 0            // [0]=phys, [1]=mu^2, [2..5]=inlet/base/top/slip
#define WS_WT1 16           // 3 x 128x132 transposed weights
#define WS_WT4 (16 + 3 * 16896)       // 16x132 (rows>=4 zero)
#define WS_V1T (WS_WT4 + 2112)        // 64x68
#define WS_V2T (WS_V1T + 4352)        // 16x68 (rows>=1 zero)

// ---- interior-kernel LDS layout (floats) ----------------------------------
#define AW_WBUF 0
#define AW_ACT0 16896
#define AW_ACT1 27456
#define AW_MUA 38016
#define AW_MUB 39104
#define AW_V1T 40192
#define AW_V2T 44544
#define AW_VC 45632   // V0(128) c0(64) c1(64)
#define AW_BIAS 45888
#define AW_OUT5 46016 // 5 streams x 16 pts x 4 outs
#define AW_MUOUT 46336
#define AW_XY 46352
#define A_SMEM_FLOATS 46384
#define A_SMEM_BYTES (A_SMEM_FLOATS * 4)

// ---- boundary-kernel LDS layout (floats) ----------------------------------
#define BW_WBUF 0
#define BW_ACT0 16896
#define BW_ACT1 19008
#define BW_BIAS 21120
#define BW_OUT 21248
#define BW_XY 21312
#define B_SMEM_FLOATS 21344
#define B_SMEM_BYTES (B_SMEM_FLOATS * 4)

// ===========================================================================
// Prep: zero accumulators, transpose/pad weights into d_ws.
// ===========================================================================
__global__ void pinn_prep(const float* __restrict__ W1, const float* __restrict__ W2,
                          const float* __restrict__ W3, const float* __restrict__ W4,
                          const float* __restrict__ V1, const float* __restrict__ V2,
                          float* __restrict__ ws) {
  int g = blockIdx.x * blockDim.x + threadIdx.x;
  int stride = gridDim.x * blockDim.x;
  if (g < 16) ws[WS_ACC + g] = 0.0f;
  for (int i = g; i < 3 * 16384; i += stride) {
    int l = i >> 14, r = i & 16383, k = r >> 7, n = r & 127;
    const float* W = (l == 0) ? W1 : (l == 1) ? W2 : W3;
    ws[WS_WT1 + l * 16896 + n * 132 + k] = W[k * 128 + n];
  }
  for (int i = g; i < 16 * 128; i += stride) {
    int n = i >> 7, k = i & 127;
    ws[WS_WT4 + n * 132 + k] = (n < 4) ? W4[k * 4 + n] : 0.0f;
  }
  for (int i = g; i < 64 * 64; i += stride) {
    int n = i >> 6, k = i & 63;
    ws[WS_V1T + n * 68 + k] = V1[k * 64 + n];
  }
  for (int i = g; i < 16 * 64; i += stride) {
    int n = i >> 6, k = i & 63;
    ws[WS_V2T + n * 68 + k] = (n == 0) ? V2[k] : 0.0f;
  }
}

// ===========================================================================
// Interior: 16 points/WG, 5 coupled streams (val, dx, dy, dxx, dyy) through
// the MLP with fused tanh-JVP algebra; mu-MLP on wave 1; Euler residuals.
// ===========================================================================
__global__ void __launch_bounds__(128) pinn_interior(
    const float* __restrict__ x_train, const float* __restrict__ W0,
    const float* __restrict__ b0, const float* __restrict__ b1,
    const float* __restrict__ b2, const float* __restrict__ b3,
    const float* __restrict__ b4, const float* __restrict__ V0,
    const float* __restrict__ c0, const float* __restrict__ c1,
    const float* __restrict__ c2, float* __restrict__ ws) {
  extern __shared__ float sm[];
  const int tid = threadIdx.x;
  const int lane = tid & 31;
  const int wave = tid >> 5;
  const int col = lane & 15;
  const int half = lane >> 4;

  // -- cooperative preload of mu-net weights + biases + point coords
  for (int i = tid * 4; i < 4352; i += 512)
    *(v4f*)(sm + AW_V1T + i) = *(const v4f*)(ws + WS_V1T + i);
  for (int i = tid * 4; i < 1088; i += 512)
    *(v4f*)(sm + AW_V2T + i) = *(const v4f*)(ws + WS_V2T + i);
  sm[AW_VC + tid] = V0[tid];
  if (tid < 64) {
    sm[AW_VC + 128 + tid] = c0[tid];
    sm[AW_VC + 192 + tid] = c1[tid];
  }
  if (tid < 16) {
    int p = blockIdx.x * 16 + tid;
    sm[AW_XY + tid * 2 + 0] = x_train[p * 2 + 0];
    sm[AW_XY + tid * 2 + 1] = x_train[p * 2 + 1];
  }
  __syncthreads();

  // -- layer 0 (K=2): direct VALU, seeds all 5 streams
  {
    float w0 = W0[tid], w1 = W0[128 + tid], bb = b0[tid];
#pragma unroll 4
    for (int p = 0; p < 16; p++) {
      float x = sm[AW_XY + p * 2], y = sm[AW_XY + p * 2 + 1];
      float z = fmaf(x, w0, fmaf(y, w1, bb));
      float a = tanh_fast(z);
      float t = 1.0f - a * a;
      float m2at = -2.0f * a * t;
      int o = AW_ACT0 + p * 132 + tid;
      sm[o] = a;
      sm[o + 2112] = t * w0;
      sm[o + 2 * 2112] = t * w1;
      sm[o + 3 * 2112] = m2at * w0 * w0;
      sm[o + 4 * 2112] = m2at * w1 * w1;
    }
  }
  __syncthreads();

  // -- layers 1..3: WMMA, 2 N-chunks/wave x 5 streams, pipelined frag loads
  int ain = AW_ACT0, aout = AW_ACT1;
  for (int l = 0; l < 3; l++) {
    const float* wsrc = ws + WS_WT1 + l * 16896;
    for (int i = tid * 4; i < 16896; i += 512)
      *(v4f*)(sm + AW_WBUF + i) = *(const v4f*)(wsrc + i);
    const float* bl = (l == 0) ? b1 : (l == 1) ? b2 : b3;
    sm[AW_BIAS + tid] = bl[tid];
    __syncthreads();

    const int nb = wave * 32;
    v8f acc[2][5];
    acc[0][0] = splat8(sm[AW_BIAS + nb + col]);
    acc[1][0] = splat8(sm[AW_BIAS + nb + 16 + col]);
#pragma unroll
    for (int j = 0; j < 2; j++)
#pragma unroll
      for (int s = 1; s < 5; s++) acc[j][s] = splat8(0.0f);

    // hoisted per-stream bases so ds offsets fold to immediates
    int ab[5];
#pragma unroll
    for (int s = 0; s < 5; s++) ab[s] = ain + col * 132 + half * 2 + s * 2112;
    const int bb0 = AW_WBUF + (nb + col) * 132 + half * 2;
    const int bb1 = bb0 + 16 * 132;

    v2f af[2][5], bfr[2][2];
#pragma unroll
    for (int s = 0; s < 5; s++) af[0][s] = *(v2f*)(sm + ab[s]);
    bfr[0][0] = *(v2f*)(sm + bb0);
    bfr[0][1] = *(v2f*)(sm + bb1);

#pragma unroll
    for (int kc = 0; kc < 32; kc++) {
      const int cur = kc & 1, nxt = cur ^ 1;
      if (kc < 31) {
        const int o = (kc + 1) * 4;
        bfr[nxt][0] = *(v2f*)(sm + bb0 + o);
        bfr[nxt][1] = *(v2f*)(sm + bb1 + o);
#pragma unroll
        for (int s = 0; s < 5; s++) af[nxt][s] = *(v2f*)(sm + ab[s] + o);
      }
#pragma unroll
      for (int s = 0; s < 5; s++) {
        acc[0][s] = wmma4(af[cur][s], bfr[cur][0], acc[0][s]);
        acc[1][s] = wmma4(af[cur][s], bfr[cur][1], acc[1][s]);
      }
    }

    // fused tanh + first/second tangent propagation
#pragma unroll
    for (int j = 0; j < 2; j++) {
      int ncol = nb + j * 16 + col;
#pragma unroll
      for (int r = 0; r < 8; r++) {
        float zv = acc[j][0][r], zx = acc[j][1][r], zy = acc[j][2][r];
        float zxx = acc[j][3][r], zyy = acc[j][4][r];
        float a = tanh_fast(zv);
        float t = 1.0f - a * a;
        float m2at = -2.0f * a * t;
        int o = aout + (r + half * 8) * 132 + ncol;
        sm[o] = a;
        sm[o + 2112] = t * zx;
        sm[o + 2 * 2112] = t * zy;
        sm[o + 3 * 2112] = fmaf(m2at * zx, zx, t * zxx);
        sm[o + 4 * 2112] = fmaf(m2at * zy, zy, t * zyy);
      }
    }
    __syncthreads();
    int tswap = ain; ain = aout; aout = tswap;
  }

  // -- layer 4 (128->4 padded to 16) + mu-MLP concurrently
  for (int i = tid * 4; i < 2112; i += 512)
    *(v4f*)(sm + AW_WBUF + i) = *(const v4f*)(ws + WS_WT4 + i);
  __syncthreads();

  if (wave == 0) {
    v8f a4[5];
    a4[0] = splat8((col < 4) ? b4[col] : 0.0f);
#pragma unroll
    for (int s = 1; s < 5; s++) a4[s] = splat8(0.0f);
    int ab[5];
#pragma unroll
    for (int s = 0; s < 5; s++) ab[s] = ain + col * 132 + half * 2 + s * 2112;
    const int bb = AW_WBUF + col * 132 + half * 2;

    v2f af[2][5], bfr[2];
#pragma unroll
    for (int s = 0; s < 5; s++) af[0][s] = *(v2f*)(sm + ab[s]);
    bfr[0] = *(v2f*)(sm + bb);
#pragma unroll
    for (int kc = 0; kc < 32; kc++) {
      const int cur = kc & 1, nxt = cur ^ 1;
      if (kc < 31) {
        const int o = (kc + 1) * 4;
        bfr[nxt] = *(v2f*)(sm + bb + o);
#pragma unroll
        for (int s = 0; s < 5; s++) af[nxt][s] = *(v2f*)(sm + ab[s] + o);
      }
#pragma unroll
      for (int s = 0; s < 5; s++) a4[s] = wmma4(af[cur][s], bfr[cur], a4[s]);
    }
    if (col < 4) {
#pragma unroll
      for (int s = 0; s < 5; s++)
#pragma unroll
        for (int r = 0; r < 8; r++)
          sm[AW_OUT5 + s * 64 + (r + half * 8) * 4 + col] = a4[s][r];
    }
  } else if (wave == 1) {
    // mu-MLP: 2 -> 64 -> 64 -> 1
    for (int i = lane; i < 1024; i += 32) {
      int p = i >> 6, n = i & 63;
      float x = sm[AW_XY + p * 2], y = sm[AW_XY + p * 2 + 1];
      float z = fmaf(x, sm[AW_VC + n], fmaf(y, sm[AW_VC + 64 + n], sm[AW_VC + 128 + n]));
      sm[AW_MUA + p * 68 + n] = tanh_fast(z);
    }
    for (int nc = 0; nc < 4; nc++) {
      int nb2 = nc * 16;
      v8f am = splat8(sm[AW_VC + 192 + nb2 + col]);
      const int ab = AW_MUA + col * 68 + half * 2;
      const int bb = AW_V1T + (nb2 + col) * 68 + half * 2;
      v2f afm[2], bfm[2];
      afm[0] = *(v2f*)(sm + ab);
      bfm[0] = *(v2f*)(sm + bb);
#pragma unroll
      for (int kc = 0; kc < 16; kc++) {
        const int cur = kc & 1, nxt = cur ^ 1;
        if (kc < 15) {
          const int o = (kc + 1) * 4;
          afm[nxt] = *(v2f*)(sm + ab + o);
          bfm[nxt] = *(v2f*)(sm + bb + o);
        }
        am = wmma4(afm[cur], bfm[cur], am);
      }
#pragma unroll
      for (int r = 0; r < 8; r++)
        sm[AW_MUB + (r + half * 8) * 68 + nb2 + col] = tanh_fast(am[r]);
    }
    v8f am = splat8(0.0f);
    const int ab = AW_MUB + col * 68 + half * 2;
    const int bb = AW_V2T + col * 68 + half * 2;
    v2f afm[2], bfm[2];
    afm[0] = *(v2f*)(sm + ab);
    bfm[0] = *(v2f*)(sm + bb);
#pragma unroll
    for (int kc = 0; kc < 16; kc++) {
      const int cur = kc & 1, nxt = cur ^ 1;
      if (kc < 15) {
        const int o = (kc + 1) * 4;
        afm[nxt] = *(v2f*)(sm + ab + o);
        bfm[nxt] = *(v2f*)(sm + bb + o);
      }
      am = wmma4(afm[cur], bfm[cur], am);
    }
    float c2v = c2[0];
    if (col == 0) {
#pragma unroll
      for (int r = 0; r < 8; r++) sm[AW_MUOUT + r + half * 8] = am[r] + c2v;
    }
  }
  __syncthreads();

  // -- per-point Euler residuals + reduction
  if (wave == 0 && lane < 16) {
    const int p = lane;
#define O5(s, j) sm[AW_OUT5 + (s) * 64 + p * 4 + (j)]
    float rho = O5(0, 0), pr = O5(0, 1), u = O5(0, 2), v = O5(0, 3);
    float rx = O5(1, 0), px = O5(1, 1), ux = O5(1, 2), vx = O5(1, 3);
    float ry = O5(2, 0), py = O5(2, 1), uy = O5(2, 2), vy = O5(2, 3);
    float rxx = O5(3, 0), pxx = O5(3, 1), uxx = O5(3, 2), vxx = O5(3, 3);
    float ryy = O5(4, 0), pyy = O5(4, 1), uyy = O5(4, 2), vyy = O5(4, 3);
#undef O5
    float o = sm[AW_MUOUT + p];
    float mu = 0.01f * o * o;
    const float ig = 2.5f;  // 1/(gamma-1)
    float q = u * u + v * v;
    float E = pr * ig + 0.5f * rho * q;
    float Ex = px * ig + 0.5f * rx * q + rho * (u * ux + v * vx);
    float Ey = py * ig + 0.5f * ry * q + rho * (u * uy + v * vy);
    float f1x = rx * u + rho * ux;
    float f2x = rx * u * u + 2.0f * rho * u * ux + px;
    float f3x = rx * u * v + rho * (ux * v + u * vx);
    float f4x = ux * (E + pr) + u * (Ex + px);
    float g1y = ry * v + rho * vy;
    float g2y = ry * u * v + rho * (uy * v + u * vy);
    float g3y = ry * v * v + 2.0f * rho * v * vy + py;
    float g4y = vy * (E + pr) + v * (Ey + py);
    float U1s = rxx + ryy;
    float U2s = (rxx * u + 2.0f * rx * ux + rho * uxx) + (ryy * u + 2.0f * ry * uy + rho * uyy);
    float U3s = (rxx * v + 2.0f * rx * vx + rho * vxx) + (ryy * v + 2.0f * ry * vy + rho * vyy);
    float qx = 2.0f * (u * ux + v * vx), qy = 2.0f * (u * uy + v * vy);
    float qxx = 2.0f * (ux * ux + u * uxx + vx * vx + v * vxx);
    float qyy = 2.0f * (uy * uy + u * uyy + vy * vy + v * vyy);
    float Exx = pxx * ig + 0.5f * (rxx * q + 2.0f * rx * qx + rho * qxx);
    float Eyy = pyy * ig + 0.5f * (ryy * q + 2.0f * ry * qy + rho * qyy);
    float U4s = Exx + Eyy;
    float r1 = f1x + g1y - mu * U1s;
    float r2 = f2x + g2y - mu * U2s;
    float r3 = f3x + g3y - mu * U3s;
    float r4 = f4x + g4y - mu * U4s;
    float phys = r1 * r1 + r2 * r2 + r3 * r3 + r4 * r4;
    float m2 = mu * mu;
    for (int off = 8; off > 0; off >>= 1) {
      phys += __shfl_down(phys, off, 16);
      m2 += __shfl_down(m2, off, 16);
    }
    if (lane == 0) {
      unsafeAtomicAdd(&ws[WS_ACC + 0], phys);
      unsafeAtomicAdd(&ws[WS_ACC + 1], m2);
    }
  }
}

// ===========================================================================
// Boundary: plain MLP forward (1 stream), 4 point sets.
// ===========================================================================
__global__ void __launch_bounds__(128) pinn_boundary(
    const float* __restrict__ x_in, const float* __restrict__ U_in,
    const float* __restrict__ x_base, const float* __restrict__ x_top,
    const float* __restrict__ x_slip, const float* __restrict__ W0,
    const float* __restrict__ b0, const float* __restrict__ b1,
    const float* __restrict__ b2, const float* __restrict__ b3,
    const float* __restrict__ b4, float* __restrict__ ws) {
  extern __shared__ float sm[];
  const int tid = threadIdx.x;
  const int lane = tid & 31;
  const int wave = tid >> 5;
  const int col = lane & 15;
  const int half = lane >> 4;
  const int set = blockIdx.x >> 8;
  const int tile = blockIdx.x & 255;
  const float* xp = (set == 0) ? x_in : (set == 1) ? x_base : (set == 2) ? x_top : x_slip;
  if (tid < 16) {
    sm[BW_XY + tid * 2 + 0] = xp[(tile * 16 + tid) * 2 + 0];
    sm[BW_XY + tid * 2 + 1] = xp[(tile * 16 + tid) * 2 + 1];
  }
  __syncthreads();

  {
    float w0 = W0[tid], w1 = W0[128 + tid], bb = b0[tid];
#pragma unroll 4
    for (int p = 0; p < 16; p++) {
      float z = fmaf(sm[BW_XY + p * 2], w0, fmaf(sm[BW_XY + p * 2 + 1], w1, bb));
      sm[BW_ACT0 + p * 132 + tid] = tanh_fast(z);
    }
  }
  __syncthreads();

  int ain = BW_ACT0, aout = BW_ACT1;
  for (int l = 0; l < 3; l++) {
    const float* wsrc = ws + WS_WT1 + l * 16896;
    for (int i = tid * 4; i < 16896; i += 512)
      *(v4f*)(sm + BW_WBUF + i) = *(const v4f*)(wsrc + i);
    const float* bl = (l == 0) ? b1 : (l == 1) ? b2 : b3;
    sm[BW_BIAS + tid] = bl[tid];
    __syncthreads();

    const int nb = wave * 32;
    v8f acc0 = splat8(sm[BW_BIAS + nb + col]);
    v8f acc1 = splat8(sm[BW_BIAS + nb + 16 + col]);
    const int ab = ain + col * 132 + half * 2;
    const int bb0 = BW_WBUF + (nb + col) * 132 + half * 2;
    const int bb1 = bb0 + 16 * 132;

    v2f af[2], bf0[2], bf1[2];
    af[0] = *(v2f*)(sm + ab);
    bf0[0] = *(v2f*)(sm + bb0);
    bf1[0] = *(v2f*)(sm + bb1);
#pragma unroll
    for (int kc = 0; kc < 32; kc++) {
      const int cur = kc & 1, nxt = cur ^ 1;
      if (kc < 31) {
        const int o = (kc + 1) * 4;
        af[nxt] = *(v2f*)(sm + ab + o);
        bf0[nxt] = *(v2f*)(sm + bb0 + o);
        bf1[nxt] = *(v2f*)(sm + bb1 + o);
      }
      acc0 = wmma4(af[cur], bf0[cur], acc0);
      acc1 = wmma4(af[cur], bf1[cur], acc1);
    }
#pragma unroll
    for (int r = 0; r < 8; r++) {
      sm[aout + (r + half * 8) * 132 + nb + col] = tanh_fast(acc0[r]);
      sm[aout + (r + half * 8) * 132 + nb + 16 + col] = tanh_fast(acc1[r]);
    }
    __syncthreads();
    int tswap = ain; ain = aout; aout = tswap;
  }

  for (int i = tid * 4; i < 2112; i += 512)
    *(v4f*)(sm + BW_WBUF + i) = *(const v4f*)(ws + WS_WT4 + i);
  __syncthreads();

  if (wave == 0) {
    v8f a4 = splat8((col < 4) ? b4[col] : 0.0f);
    const int ab = ain + col * 132 + half * 2;
    const int bb = BW_WBUF + col * 132 + half * 2;
    v2f af[2], bf[2];
    af[0] = *(v2f*)(sm + ab);
    bf[0] = *(v2f*)(sm + bb);
#pragma unroll
    for (int kc = 0; kc < 32; kc++) {
      const int cur = kc & 1, nxt = cur ^ 1;
      if (kc < 31) {
        const int o = (kc + 1) * 4;
        af[nxt] = *(v2f*)(sm + ab + o);
        bf[nxt] = *(v2f*)(sm + bb + o);
      }
      a4 = wmma4(af[cur], bf[cur], a4);
    }
    if (col < 4) {
#pragma unroll
      for (int r = 0; r < 8; r++) sm[BW_OUT + (r + half * 8) * 4 + col] = a4[r];
    }
  }
  __syncthreads();

  if (wave == 0 && lane < 16) {
    const int p = lane;
    float o0 = sm[BW_OUT + p * 4 + 0], o1 = sm[BW_OUT + p * 4 + 1];
    float o2 = sm[BW_OUT + p * 4 + 2], o3 = sm[BW_OUT + p * 4 + 3];
    float loss;
    if (set == 0) {
      int gi = (tile * 16 + p) * 4;
      float d0 = o0 - U_in[gi + 0], d1 = o1 - U_in[gi + 1];
      float d2 = o2 - U_in[gi + 2], d3 = o3 - U_in[gi + 3];
      loss = d0 * d0 + d1 * d1 + d2 * d2 + d3 * d3;
    } else if (set == 3) {
      float s_ = 0.17364817766f * o2 + 0.98480775301f * o3;  // -sin(-pi/18), cos
      loss = s_ * s_;
    } else {
      loss = o3 * o3;
    }
    for (int off = 8; off > 0; off >>= 1) loss += __shfl_down(loss, off, 16);
    if (lane == 0) unsafeAtomicAdd(&ws[WS_ACC + 2 + set], loss);
  }
}

// ===========================================================================
__global__ void pinn_final(const float* __restrict__ ws, float* __restrict__ out) {
  if (threadIdx.x == 0 && blockIdx.x == 0) {
    float phys = ws[WS_ACC + 0] * (1.0f / 32768.0f);
    float mu2 = ws[WS_ACC + 1] * (1.0f / 32768.0f);
    float bnd = (ws[WS_ACC + 2] + ws[WS_ACC + 3] + ws[WS_ACC + 4] + ws[WS_ACC + 5]) *
                (1.0f / 4096.0f);
    out[0] = phys + 10.0f * bnd + 0.1f * mu2;
  }
}

// ===========================================================================
extern "C" void kernel_launch(void* const* d_in, const int* in_sizes, int n_in,
                              void* d_out, int out_size, void* d_ws, size_t ws_size,
                              hipStream_t stream) {
  (void)in_sizes; (void)n_in; (void)out_size; (void)ws_size;
  const float* x_train = (const float*)d_in[0];
  const float* x_inlet = (const float*)d_in[1];
  const float* U_inlet = (const float*)d_in[2];
  const float* x_base = (const float*)d_in[3];
  const float* x_top = (const float*)d_in[4];
  const float* x_slip = (const float*)d_in[5];
  const float* W0 = (const float*)d_in[6];
  const float* b0 = (const float*)d_in[7];
  const float* W1 = (const float*)d_in[8];
  const float* b1 = (const float*)d_in[9];
  const float* W2 = (const float*)d_in[10];
  const float* b2 = (const float*)d_in[11];
  const float* W3 = (const float*)d_in[12];
  const float* b3 = (const float*)d_in[13];
  const float* W4 = (const float*)d_in[14];
  const float* b4 = (const float*)d_in[15];
  const float* V0 = (const float*)d_in[16];
  const float* c0 = (const float*)d_in[17];
  const float* V1 = (const float*)d_in[18];
  const float* c1 = (const float*)d_in[19];
  const float* V2 = (const float*)d_in[20];
  const float* c2 = (const float*)d_in[21];
  float* ws = (float*)d_ws;
  float* out = (float*)d_out;

  (void)hipFuncSetAttribute((const void*)pinn_interior,
                            hipFuncAttributeMaxDynamicSharedMemorySize, A_SMEM_BYTES);
  (void)hipFuncSetAttribute((const void*)pinn_boundary,
                            hipFuncAttributeMaxDynamicSharedMemorySize, B_SMEM_BYTES);

  pinn_prep<<<64, 256, 0, stream>>>(W1, W2, W3, W4, V1, V2, ws);
  pinn_interior<<<2048, 128, A_SMEM_BYTES, stream>>>(x_train, W0, b0, b1, b2, b3,
                                                     b4, V0, c0, c1, c2, ws);
  pinn_boundary<<<1024, 128, B_SMEM_BYTES, stream>>>(x_inlet, U_inlet, x_base,
                                                     x_top, x_slip, W0, b0, b1,
                                                     b2, b3, b4, ws);
  pinn_final<<<1, 32, 0, stream>>>(ws, out);
}